// GeneralizedMaxwellNetwork_87162066305736
// MI455X (gfx1250) — compile-verified
//
#include <hip/hip_runtime.h>

#define DT_C   0.1f
#define NB     256
#define NT     4096
#define NE     64
#define LCH    16
#define NCH    (NT / LCH)      // 256 chunks
#define NROWS  (NB * NCH)      // 65536 (b,chunk) rows

typedef float v2f __attribute__((ext_vector_type(2)));
typedef float v8f __attribute__((ext_vector_type(8)));

// workspace layout (float offsets)
#define OFF_E     0                       // [64]   E = exp(log_E)
#define OFF_ETA   64                      // [64]   eta
#define OFF_DL    128                     // [64]   decay^16
#define OFF_P     192                     // [64*16] P[j*16+m] = decay_j^m
#define OFF_MEAN  1216                    // [256*2] time-mean of x
#define OFF_COEF  1728                    // [256*64] element_weights * gate
#define OFF_W     18112                   // [256*16] FIR taps W[b,m]
#define OFF_C     22208                   // [65536*64] chunk injections
#define OFF_SC    (22208 + 4194304)       // [65536*64] chunk-start states

static __device__ __forceinline__ v8f wmma_f32_16x16x4(v2f a, v2f b, v8f c) {
  return __builtin_amdgcn_wmma_f32_16x16x4_f32(false, a, false, b, (short)0, c,
                                               false, false);
}

// ---------------------------------------------------------------- params ----
__global__ void k_params(const float* __restrict__ logE,
                         const float* __restrict__ logEta,
                         float* __restrict__ ws) {
  int j = threadIdx.x;
  if (j < NE) {
    float E     = __expf(logE[j]);
    float eta   = __expf(logEta[j]);
    float decay = __expf(-DT_C * E / eta);   // exp(-DT/(eta/E))
    ws[OFF_E + j]   = E;
    ws[OFF_ETA + j] = eta;
    float p = 1.0f;
    #pragma unroll
    for (int m = 0; m < LCH; ++m) { ws[OFF_P + j * LCH + m] = p; p *= decay; }
    ws[OFF_DL + j] = p;                      // decay^16
  }
}

// ------------------------------------------------------------- time mean ----
__global__ void k_mean(const float* __restrict__ x, float* __restrict__ ws) {
  int b = blockIdx.x, tid = threadIdx.x;
  const float* xb = x + (size_t)b * NT * 2;
  float s0 = 0.f, s1 = 0.f;
  for (int t = tid; t < NT; t += blockDim.x) {
    s0 += xb[2 * t];
    s1 += xb[2 * t + 1];
  }
  __shared__ float r0[256], r1[256];
  r0[tid] = s0; r1[tid] = s1;
  __syncthreads();
  for (int off = 128; off > 0; off >>= 1) {
    if (tid < off) { r0[tid] += r0[tid + off]; r1[tid] += r1[tid + off]; }
    __syncthreads();
  }
  if (tid == 0) {
    ws[OFF_MEAN + b * 2 + 0] = r0[0] * (1.0f / NT);
    ws[OFF_MEAN + b * 2 + 1] = r1[0] * (1.0f / NT);
  }
}

// ------------------------------------------------------- gate -> coef -------
__global__ void k_coef(const float* __restrict__ gw, const float* __restrict__ gb,
                       const float* __restrict__ ew, float* __restrict__ ws) {
  int idx = blockIdx.x * blockDim.x + threadIdx.x;  // 0..16383
  int b = idx >> 6, j = idx & 63;
  float m0 = ws[OFF_MEAN + b * 2 + 0];
  float m1 = ws[OFF_MEAN + b * 2 + 1];
  float z = m0 * gw[j] + m1 * gw[NE + j] + gb[j];
  float g = 1.0f / (1.0f + __expf(-z));
  ws[OFF_COEF + idx] = ew[j] * g;
}

// ----------------------------- W[b,m] = sum_j coef*E*decay^m  (WMMA GEMM) ---
// M=256 (16 tiles), N=16, K=64. One wave per M-tile.
__global__ void k_wfir(float* __restrict__ ws) {
  int l = threadIdx.x, g = l >> 4, col = l & 15;
  int row = blockIdx.x * 16 + (l & 15);
  const float* coef = ws + OFF_COEF;
  const float* E    = ws + OFF_E;
  const float* P    = ws + OFF_P;
  v8f acc = {};
  #pragma unroll
  for (int k0 = 0; k0 < NE; k0 += 4) {
    int ka = k0 + 2 * g;
    v2f a, vb;
    a.x  = coef[row * NE + ka]     * E[ka];
    a.y  = coef[row * NE + ka + 1] * E[ka + 1];
    vb.x = P[ka * LCH + col];
    vb.y = P[(ka + 1) * LCH + col];
    acc = wmma_f32_16x16x4(a, vb, acc);
  }
  float* W = ws + OFF_W;
  #pragma unroll
  for (int r = 0; r < 8; ++r)
    W[(blockIdx.x * 16 + r + 8 * g) * LCH + col] = acc[r];
}

// --------- c[b,n,j] = sum_{q=0..15} P[j,15-q] * d[b, n*16+q+1]  (WMMA) ------
// M=65536 rows (b,n), K=16, N=64. One wave per 16x64 tile.
__global__ void k_cgemm(const float* __restrict__ x, float* __restrict__ ws) {
  int l = threadIdx.x, g = l >> 4, col = l & 15;
  int tile = blockIdx.x;
  int row  = tile * 16 + (l & 15);        // r = b*NCH + n
  int b = row >> 8, n = row & (NCH - 1);
  const float* xb = x + (size_t)b * NT * 2;
  int t0 = n * LCH;
  const float* P = ws + OFF_P;
  float* c = ws + OFF_C;

  auto dval = [&](int q) -> float {       // d[b, t0+q+1], zero past end
    int t1 = t0 + q + 1;
    return (t1 < NT) ? (xb[2 * t1] - xb[2 * t1 - 2]) : 0.0f;
  };

  v8f acc[4] = {};
  #pragma unroll
  for (int k0 = 0; k0 < LCH; k0 += 4) {
    int q = k0 + 2 * g;
    v2f a;
    a.x = dval(q);
    a.y = dval(q + 1);
    #pragma unroll
    for (int nt = 0; nt < 4; ++nt) {
      int j = nt * 16 + col;
      v2f vb;
      vb.x = P[j * LCH + (15 - q)];
      vb.y = P[j * LCH + (14 - q)];
      acc[nt] = wmma_f32_16x16x4(a, vb, acc[nt]);
    }
  }
  #pragma unroll
  for (int nt = 0; nt < 4; ++nt)
    #pragma unroll
    for (int r = 0; r < 8; ++r)
      c[(size_t)(tile * 16 + r + 8 * g) * NE + nt * 16 + col] = acc[nt][r];
}

// ------------------- chunk-level scan over 256 chunks (16K parallel) --------
__global__ void k_scan(const float* __restrict__ x, float* __restrict__ ws) {
  int idx = blockIdx.x * blockDim.x + threadIdx.x;  // (b,j)
  int b = idx >> 6, j = idx & 63;
  const float* xb = x + (size_t)b * NT * 2;
  float E  = ws[OFF_E + j];
  float et = ws[OFF_ETA + j];
  float dL = ws[OFF_DL + j];
  const float* c  = ws + OFF_C;
  float*       sc = ws + OFF_SC;
  size_t base = (size_t)b * NCH * NE + j;
  float s = xb[0] * E + xb[1] * et;                 // s0
  sc[base] = s;
  for (int n = 0; n < NCH - 1; ++n) {
    __builtin_prefetch(&c[base + (size_t)(n + 8) * NE], 0, 0);
    float cv = c[base + (size_t)n * NE];
    s = dL * s + E * cv;
    sc[base + (size_t)(n + 1) * NE] = s;
  }
}

// --- out[b, n*16+i] = sum_j (coef*sc)·P[j,i]  (WMMA)  +  FIR(W, d) ----------
__global__ void k_out(const float* __restrict__ x, const float* __restrict__ ws,
                      float* __restrict__ out) {
  int l = threadIdx.x, g = l >> 4, col = l & 15;    // col == output step i
  int tile = blockIdx.x;
  int row  = tile * 16 + (l & 15);                  // r = b*NCH + n
  int b = row >> 8;                                 // all rows in tile share b
  const float* coef = ws + OFF_COEF;
  const float* sc   = ws + OFF_SC;
  const float* P    = ws + OFF_P;
  v8f acc = {};
  #pragma unroll
  for (int k0 = 0; k0 < NE; k0 += 4) {
    int ka = k0 + 2 * g;
    v2f a, vb;
    a.x  = coef[b * NE + ka]     * sc[(size_t)row * NE + ka];
    a.y  = coef[b * NE + ka + 1] * sc[(size_t)row * NE + ka + 1];
    vb.x = P[ka * LCH + col];
    vb.y = P[(ka + 1) * LCH + col];
    acc = wmma_f32_16x16x4(a, vb, acc);
  }
  // FIR correction + store; D layout: VGPR r -> row (r + 8g) of tile, col = i
  const float* W  = ws + OFF_W;
  const float* xb = x + (size_t)b * NT * 2;
  int i = col;
  #pragma unroll
  for (int r = 0; r < 8; ++r) {
    int rowm = tile * 16 + r + 8 * g;
    int n = rowm & (NCH - 1);
    int t0 = n * LCH;
    float fir = 0.0f;
    for (int q = 0; q < i; ++q) {
      float d = xb[2 * (t0 + q + 1)] - xb[2 * (t0 + q)];
      fir += W[b * LCH + (i - 1 - q)] * d;
    }
    out[(size_t)b * NT + t0 + i] = acc[r] + fir;
  }
}

// ---------------------------------------------------------------------------
extern "C" void kernel_launch(void* const* d_in, const int* in_sizes, int n_in,
                              void* d_out, int out_size, void* d_ws, size_t ws_size,
                              hipStream_t stream) {
  const float* x    = (const float*)d_in[0];   // (256, 4096, 2)
  const float* logE = (const float*)d_in[1];   // (64,)
  const float* logN = (const float*)d_in[2];   // (64,)
  const float* ew   = (const float*)d_in[3];   // (64,)
  const float* gw   = (const float*)d_in[4];   // (2, 64)
  const float* gb   = (const float*)d_in[5];   // (64,)
  float* out = (float*)d_out;                  // (256, 4096)
  float* ws  = (float*)d_ws;                   // needs ~33.7 MB

  k_params<<<1, 64, 0, stream>>>(logE, logN, ws);
  k_mean  <<<NB, 256, 0, stream>>>(x, ws);
  k_coef  <<<NB * NE / 256, 256, 0, stream>>>(gw, gb, ew, ws);
  k_wfir  <<<NB / 16, 32, 0, stream>>>(ws);
  k_cgemm <<<NROWS / 16, 32, 0, stream>>>(x, ws);
  k_scan  <<<NB * NE / 256, 256, 0, stream>>>(x, ws);
  k_out   <<<NROWS / 16, 32, 0, stream>>>(x, ws, out);
}